// LightGCN_65506841198657
// MI455X (gfx1250) — compile-verified
//
#include <hip/hip_runtime.h>
#include <stdint.h>

#define DIM 64
#define CHUNK 256   // edges staged per block iteration

#ifndef __has_builtin
#define __has_builtin(x) 0
#endif

#if __has_builtin(__builtin_amdgcn_global_load_async_to_lds_b128)
#define USE_ASYNC_LDS 1
#else
#define USE_ASYNC_LDS 0
#endif

#if USE_ASYNC_LDS
typedef int v4i_t __attribute__((vector_size(16)));
typedef __attribute__((address_space(1))) v4i_t* gv4i_p;   // global int4*
typedef __attribute__((address_space(3))) v4i_t* lv4i_p;   // LDS int4*
// CPol: TH[2:0]=1 (non-temporal) -- edge stream is touched once per hop and
// must not evict the L2-resident gather/scatter buffers (192 MB L2 vs ~250 MB
// total working set if the stream were cached).
#define ASYNC_CPOL 1
#define WAIT_ASYNC_1() asm volatile("s_wait_asynccnt 1" ::: "memory")
#define WAIT_ASYNC_0() asm volatile("s_wait_asynccnt 0" ::: "memory")
#else
#define WAIT_ASYNC_1()
#define WAIT_ASYNC_0()
#endif

// Stage CHUNK edge triples (rows/cols/vals) into LDS.
// Async path: waves 0-5 each issue exactly one b128 async-to-LDS (16B/lane),
// waves 6-7 issue nothing.  Sync fallback: each thread copies one element.
__device__ __forceinline__ void stage_chunk(const int* __restrict__ rows,
                                            const int* __restrict__ cols,
                                            const float* __restrict__ vals,
                                            unsigned base, int tid,
                                            int* sr, int* sc, float* sv) {
#if USE_ASYNC_LDS
  if (tid < 192) {
    const void* g;
    void* l;
    if (tid < 64) {
      g = rows + base + tid * 4;
      l = sr + tid * 4;
    } else if (tid < 128) {
      g = cols + base + (tid - 64) * 4;
      l = sc + (tid - 64) * 4;
    } else {
      g = vals + base + (tid - 128) * 4;
      l = sv + (tid - 128) * 4;
    }
    __builtin_amdgcn_global_load_async_to_lds_b128((gv4i_p)g, (lv4i_p)l, 0,
                                                   ASYNC_CPOL);
  }
#else
  sr[tid] = rows[base + tid];
  sc[tid] = cols[base + tid];
  sv[tid] = vals[base + tid];
#endif
}

// Edge-parallel COO SpMM: dst[r,:] += v * src[c,:].
// 16 lanes per edge, one float4 per lane -> each edge's 256B row is a fully
// coalesced 16-lane access.  dst (76.8 MB) and src are L2-resident on MI455X
// (192 MB L2), so the f32 atomics resolve at L2 speed.
__global__ __launch_bounds__(256) void k_spmm(
    const int* __restrict__ rows, const int* __restrict__ cols,
    const float* __restrict__ vals, const float* __restrict__ srcU,
    const float* __restrict__ srcI, float* __restrict__ dst, int n_users,
    int n_chunks) {
  __shared__ __align__(16) int s_rows[2][CHUNK];
  __shared__ __align__(16) int s_cols[2][CHUNK];
  __shared__ __align__(16) float s_vals[2][CHUNK];

  const int tid = threadIdx.x;
  const int q4 = (tid & 15) * 4;  // float offset within the 64-float row
  const int grp = tid >> 4;       // which of 16 edges this thread serves
  const unsigned stride = gridDim.x;

  unsigned chunk = blockIdx.x;
  stage_chunk(rows, cols, vals, chunk * (unsigned)CHUNK, tid, s_rows[0],
              s_cols[0], s_vals[0]);
  int buf = 0;

  while (true) {
    const unsigned nxt = chunk + stride;
    const bool has_next = nxt < (unsigned)n_chunks;
    if (has_next) {
      stage_chunk(rows, cols, vals, nxt * (unsigned)CHUNK, tid,
                  s_rows[buf ^ 1], s_cols[buf ^ 1], s_vals[buf ^ 1]);
      WAIT_ASYNC_1();  // current buffer complete; next may stay in flight
    } else {
      WAIT_ASYNC_0();
    }
    __syncthreads();

#pragma unroll
    for (int it = 0; it < 16; ++it) {
      const int el = (it << 4) + grp;
      const int c = s_cols[buf][el];
      const int r = s_rows[buf][el];
      const float v = s_vals[buf][el];
      const float* s = (c < n_users)
                           ? (srcU + (size_t)c * DIM + q4)
                           : (srcI + (size_t)(c - n_users) * DIM + q4);
      const float4 x = *(const float4*)s;
      float* d = dst + (size_t)r * DIM + q4;
      atomicAdd(d + 0, v * x.x);
      atomicAdd(d + 1, v * x.y);
      atomicAdd(d + 2, v * x.z);
      atomicAdd(d + 3, v * x.w);
    }

    if (!has_next) break;
    __syncthreads();  // everyone done reading buf before it is re-staged
    buf ^= 1;
    chunk = nxt;
  }
}

// Remainder edges (n_edges % CHUNK) -- direct, unstaged.
__global__ void k_spmm_tail(const int* __restrict__ rows,
                            const int* __restrict__ cols,
                            const float* __restrict__ vals,
                            const float* __restrict__ srcU,
                            const float* __restrict__ srcI,
                            float* __restrict__ dst, int n_users, int e0,
                            int n_edges) {
  const unsigned g = blockIdx.x * 256u + threadIdx.x;
  const int e = e0 + (int)(g >> 4);
  if (e >= n_edges) return;
  const int q4 = (int)(g & 15u) * 4;
  const int c = __builtin_nontemporal_load(cols + e);
  const int r = __builtin_nontemporal_load(rows + e);
  const float v = __builtin_nontemporal_load(vals + e);
  const float* s = (c < n_users) ? (srcU + (size_t)c * DIM + q4)
                                 : (srcI + (size_t)(c - n_users) * DIM + q4);
  const float4 x = *(const float4*)s;
  float* d = dst + (size_t)r * DIM + q4;
  atomicAdd(d + 0, v * x.x);
  atomicAdd(d + 1, v * x.y);
  atomicAdd(d + 2, v * x.z);
  atomicAdd(d + 3, v * x.w);
}

__global__ void k_zero(float4* __restrict__ p, size_t n4) {
  const size_t i = (size_t)blockIdx.x * blockDim.x + threadIdx.x;
  if (i < n4) p[i] = make_float4(0.f, 0.f, 0.f, 0.f);
}

// out = (emb + h1 + out) / 3   (out currently holds h2)
__global__ void k_final(const float4* __restrict__ uemb,
                        const float4* __restrict__ iemb,
                        const float4* __restrict__ h1,
                        float4* __restrict__ out, size_t u4, size_t n4) {
  const size_t i = (size_t)blockIdx.x * blockDim.x + threadIdx.x;
  if (i >= n4) return;
  const float4 e = (i < u4) ? uemb[i] : iemb[i - u4];
  const float4 a = h1[i];
  const float4 b = out[i];
  const float s = 1.0f / 3.0f;
  float4 o;
  o.x = (e.x + a.x + b.x) * s;
  o.y = (e.y + a.y + b.y) * s;
  o.z = (e.z + a.z + b.z) * s;
  o.w = (e.w + a.w + b.w) * s;
  out[i] = o;
}

extern "C" void kernel_launch(void* const* d_in, const int* in_sizes, int n_in,
                              void* d_out, int out_size, void* d_ws,
                              size_t ws_size, hipStream_t stream) {
  const int* rows = (const int*)d_in[0];
  const int* cols = (const int*)d_in[1];
  const float* vals = (const float*)d_in[2];
  const float* uemb = (const float*)d_in[3];
  const float* iemb = (const float*)d_in[4];
  float* out = (float*)d_out;

  const int n_edges = in_sizes[0];
  const int n_users = in_sizes[3] / DIM;
  const int n_items = in_sizes[4] / DIM;
  const int n_nodes = n_users + n_items;
  const size_t nf = (size_t)n_nodes * DIM;  // total node-features
  const size_t n4 = nf / 4;

  float* h1 = (float*)d_ws;  // hop-1 result, nf floats (76.8 MB)

  // Zero both accumulators (d_out serves as the h2 accumulator).
  {
    const int grid = (int)((n4 + 255) / 256);
    k_zero<<<grid, 256, 0, stream>>>((float4*)h1, n4);
    k_zero<<<grid, 256, 0, stream>>>((float4*)out, n4);
  }

  const int n_chunks = n_edges / CHUNK;
  const int rem = n_edges - n_chunks * CHUNK;
  const int spmm_grid = (n_chunks < 4096) ? n_chunks : 4096;

  // Hop 1: h1 = A @ [uemb; iemb]
  if (n_chunks > 0)
    k_spmm<<<spmm_grid, 256, 0, stream>>>(rows, cols, vals, uemb, iemb, h1,
                                          n_users, n_chunks);
  if (rem > 0) {
    const int grid = (rem * 16 + 255) / 256;
    k_spmm_tail<<<grid, 256, 0, stream>>>(rows, cols, vals, uemb, iemb, h1,
                                          n_users, n_chunks * CHUNK, n_edges);
  }

  // Hop 2: out(h2) = A @ h1   (src viewed as one contiguous node array)
  const float* h1u = h1;
  const float* h1i = h1 + (size_t)n_users * DIM;
  if (n_chunks > 0)
    k_spmm<<<spmm_grid, 256, 0, stream>>>(rows, cols, vals, h1u, h1i, out,
                                          n_users, n_chunks);
  if (rem > 0) {
    const int grid = (rem * 16 + 255) / 256;
    k_spmm_tail<<<grid, 256, 0, stream>>>(rows, cols, vals, h1u, h1i, out,
                                          n_users, n_chunks * CHUNK, n_edges);
  }

  // Finalize: out = (emb + h1 + out) / 3
  {
    const int grid = (int)((n4 + 255) / 256);
    k_final<<<grid, 256, 0, stream>>>((const float4*)uemb, (const float4*)iemb,
                                      (const float4*)h1, (float4*)out,
                                      (size_t)n_users * (DIM / 4), n4);
  }
}